// SubgraphGNN_46591805227284
// MI455X (gfx1250) — compile-verified
//
#include <hip/hip_runtime.h>
#include <hip/hip_bf16.h>

typedef float v2f __attribute__((ext_vector_type(2)));
typedef float v8f __attribute__((ext_vector_type(8)));

#define D_FEAT 128

// ---------------- degree / norm precompute (shared across all 3 layers) ----
__global__ __launch_bounds__(256) void deg_kernel(const long long* __restrict__ ei,
                                                  const float* __restrict__ ew,
                                                  float* __restrict__ deg,
                                                  int E, int N) {
    long long e = (long long)blockIdx.x * blockDim.x + threadIdx.x;
    if (e >= (long long)E + N) return;
    if (e < E) {
        atomicAdd(&deg[(int)ei[E + e]], ew[e]);  // dst-indexed
    } else {
        atomicAdd(&deg[(int)(e - E)], 1.0f);     // self loop weight 1
    }
}

__global__ __launch_bounds__(256) void dinv_kernel(float* __restrict__ deg, int N) {
    int i = blockIdx.x * blockDim.x + threadIdx.x;
    if (i >= N) return;
    float d = deg[i];
    deg[i] = (d > 0.0f) ? rsqrtf(d) : 0.0f;      // in place: deg -> dinv
}

__global__ __launch_bounds__(256) void norm_kernel(const long long* __restrict__ ei,
                                                   const float* __restrict__ ew,
                                                   const float* __restrict__ dinv,
                                                   float* __restrict__ nrm,
                                                   int E, int N) {
    long long e = (long long)blockIdx.x * blockDim.x + threadIdx.x;
    if (e >= (long long)E + N) return;
    if (e < E) {
        nrm[e] = dinv[(int)ei[e]] * ew[e] * dinv[(int)ei[E + e]];
    } else {
        float di = dinv[(int)(e - E)];
        nrm[e] = di * di;
    }
}

// ---------------- GEMM: Y[N,128] = X[N,128] @ W[128,128] via f32 WMMA -------
// 8 waves / block; wave w computes the 16x16 tile (rows blockIdx*16, cols w*16).
// K=4 per V_WMMA_F32_16X16X4_F32, fully unrolled -> 32 WMMAs per tile.
// W staged in LDS (64KB of the 320KB WGP pool).
__global__ __launch_bounds__(256) void gemm128_wmma(const float* __restrict__ X,
                                                    const float* __restrict__ W,
                                                    float* __restrict__ Y,
                                                    int N) {
    __shared__ float sW[D_FEAT * D_FEAT];
    for (int i = threadIdx.x; i < D_FEAT * D_FEAT; i += 256)
        sW[i] = W[i];
    __syncthreads();

    const int wave = threadIdx.x >> 5;           // 0..7 -> column tile
    const int lane = threadIdx.x & 31;
    const int half = lane >> 4;                  // selects K-pair within quad
    const int lr   = lane & 15;
    const int row0 = blockIdx.x * 16;
    const int n0 = wave * 16;

    // clamp row reads (last partial tile only)
    int rrow = row0 + lr;
    if (rrow >= N) rrow = N - 1;
    const float* xrow = X + (size_t)rrow * D_FEAT + 2 * half;  // 8B aligned
    const float* wbase = sW + 2 * half * D_FEAT + n0 + lr;

    v8f c = {0.f, 0.f, 0.f, 0.f, 0.f, 0.f, 0.f, 0.f};
#pragma unroll
    for (int k0 = 0; k0 < D_FEAT; k0 += 4) {
        // A 16x4 f32: lanes 0-15 hold M, VGPR0/1 = K pair (half selects K+2)
        v2f a = *(const v2f*)(xrow + k0);
        // B 4x16 f32: row (K) striped across VGPR/halves, col = lane
        v2f b;
        b.x = wbase[k0 * D_FEAT];
        b.y = wbase[k0 * D_FEAT + D_FEAT];
        c = __builtin_amdgcn_wmma_f32_16x16x4_f32(false, a, false, b,
                                                  (short)0, c, false, false);
    }

    // C/D layout: VGPR v -> row (row0 + v + 8*half), col = n0 + (lane&15)
    if (row0 + 16 <= N) {                        // fast path: whole tile in range
        float* yp = Y + (size_t)(row0 + 8 * half) * D_FEAT + n0 + lr;
#pragma unroll
        for (int v = 0; v < 8; ++v)
            yp[v * D_FEAT] = c[v];
    } else {                                     // final partial tile only
#pragma unroll
        for (int v = 0; v < 8; ++v) {
            int row = row0 + v + 8 * half;
            if (row < N) Y[(size_t)row * D_FEAT + n0 + lr] = c[v];
        }
    }
}

// ---------------- edge scatter: agg[dst] += hlin[src] * norm ----------------
__global__ __launch_bounds__(128) void scatter_kernel(const float* __restrict__ hlin,
                                                      const long long* __restrict__ ei,
                                                      const float* __restrict__ nrm,
                                                      float* __restrict__ agg,
                                                      int E) {
    const long long e = blockIdx.x;              // one edge per 128-thread block
    const int t = threadIdx.x;
    int s, d;
    if (e < E) { s = (int)ei[e]; d = (int)ei[E + e]; }
    else       { s = d = (int)(e - E); }         // self loop
    const float w = nrm[e];
    atomicAdd(&agg[d * D_FEAT + t], hlin[s * D_FEAT + t] * w);
}

// ---------------- per-feature sum / sumsq for BN ----------------------------
__global__ __launch_bounds__(128) void stats_kernel(const float* __restrict__ agg,
                                                    float* __restrict__ stats,
                                                    int N) {
    const int t = threadIdx.x;                   // feature
    int r0 = blockIdx.x * 64;
    int r1 = r0 + 64; if (r1 > N) r1 = N;
    float s = 0.f, sq = 0.f;
    for (int r = r0; r < r1; ++r) {
        float v = agg[(size_t)r * D_FEAT + t];
        s += v; sq += v * v;
    }
    atomicAdd(&stats[t], s);
    atomicAdd(&stats[D_FEAT + t], sq);
}

// ---------------- fold BN into per-feature scale/shift ----------------------
// (post-aggregation bias cancels exactly under mean-centering; dropped)
__global__ __launch_bounds__(128) void finalize_stats_kernel(const float* __restrict__ stats,
                                                             const float* __restrict__ g,
                                                             const float* __restrict__ be,
                                                             float* __restrict__ sc,
                                                             int N) {
    const int f = threadIdx.x;
    const float inv_n = 1.0f / (float)N;
    const float mu  = stats[f] * inv_n;
    const float var = stats[D_FEAT + f] * inv_n - mu * mu;
    const float s = rsqrtf(var + 1e-5f) * g[f];
    sc[f]          = s;                          // scale
    sc[D_FEAT + f] = be[f] - mu * s;             // shift
}

// ---------------- BN + ReLU apply: one FMA + max per element ----------------
__global__ __launch_bounds__(256) void bn_relu_kernel(const float* __restrict__ agg,
                                                      const float* __restrict__ sc,
                                                      float* __restrict__ hout,
                                                      int N) {
    long long i = (long long)blockIdx.x * blockDim.x + threadIdx.x;
    if (i >= (long long)N * D_FEAT) return;
    const int f = (int)(i & (D_FEAT - 1));
    float v = fmaf(agg[i], sc[f], sc[D_FEAT + f]);
    hout[i] = v > 0.f ? v : 0.f;
}

// ---------------- masked mean pooling over subgraphs ------------------------
__global__ __launch_bounds__(128) void pool_kernel(const float* __restrict__ h,
                                                   const long long* __restrict__ sg,
                                                   float* __restrict__ emb,
                                                   int N) {
    const int s = blockIdx.x;
    const int f = threadIdx.x;
    float sum = 0.f;
    int cnt = 0;
#pragma unroll 4
    for (int m = 0; m < 32; ++m) {
        long long idx = sg[(long long)s * 32 + m];
        if (idx != -1) {
            int ci = (int)idx; if (ci < 0) ci = 0; if (ci >= N) ci = N - 1;
            sum += h[(size_t)ci * D_FEAT + f];
            cnt++;
        }
    }
    float c = (float)cnt; if (c < 1.f) c = 1.f;
    emb[(size_t)s * D_FEAT + f] = sum / c;
}

// ---------------- classifier MLP 128 -> 64 -> 10 ----------------------------
__global__ __launch_bounds__(64) void mlp_kernel(const float* __restrict__ emb,
                                                 const float* __restrict__ Wc1,
                                                 const float* __restrict__ bc1,
                                                 const float* __restrict__ Wc2,
                                                 const float* __restrict__ bc2,
                                                 float* __restrict__ out) {
    __shared__ float se[D_FEAT];
    __shared__ float sh[64];
    const int s = blockIdx.x;
    const int t = threadIdx.x;
    se[t]      = emb[(size_t)s * D_FEAT + t];
    se[64 + t] = emb[(size_t)s * D_FEAT + 64 + t];
    __syncthreads();
    float acc = bc1[t];
#pragma unroll 8
    for (int k = 0; k < D_FEAT; ++k) acc += se[k] * Wc1[k * 64 + t];
    sh[t] = acc > 0.f ? acc : 0.f;
    __syncthreads();
    if (t < 10) {
        float o = bc2[t];
#pragma unroll 8
        for (int k = 0; k < 64; ++k) o += sh[k] * Wc2[k * 10 + t];
        out[(size_t)s * 10 + t] = o;
    }
}

// ============================================================================
extern "C" void kernel_launch(void* const* d_in, const int* in_sizes, int n_in,
                              void* d_out, int out_size, void* d_ws, size_t ws_size,
                              hipStream_t stream) {
    const float*     x   = (const float*)d_in[0];
    const long long* ei  = (const long long*)d_in[1];   // [2,E] int64
    const float*     ew  = (const float*)d_in[2];
    const long long* sg  = (const long long*)d_in[3];   // [S,32] int64
    const float*     Wl[3] = { (const float*)d_in[4], (const float*)d_in[6], (const float*)d_in[8] };
    const float*     gl[3] = { (const float*)d_in[10], (const float*)d_in[12], (const float*)d_in[14] };
    const float*     bl[3] = { (const float*)d_in[11], (const float*)d_in[13], (const float*)d_in[15] };
    const float*     Wc1 = (const float*)d_in[16];
    const float*     bc1 = (const float*)d_in[17];
    const float*     Wc2 = (const float*)d_in[18];
    const float*     bc2 = (const float*)d_in[19];
    float*           out = (float*)d_out;

    const int N = in_sizes[0] / D_FEAT;      // 100000
    const int E = in_sizes[1] / 2;           // 1600000
    const int S = in_sizes[3] / 32;          // 20000
    const long long ET = (long long)E + N;   // edges + self loops

    // ---- workspace layout (256B-aligned slices) ----
    char* ws = (char*)d_ws;
    size_t off = 0;
    auto alloc = [&](size_t bytes) -> float* {
        float* p = (float*)(ws + off);
        off += (bytes + 255) & ~(size_t)255;
        return p;
    };
    float* dinv  = alloc((size_t)N * 4);                 // deg -> dinv (in place)
    float* nrm   = alloc((size_t)ET * 4);
    float* hlin  = alloc((size_t)N * D_FEAT * 4);
    float* hbuf  = alloc((size_t)N * D_FEAT * 4);        // activation ping buffer
    float* agg   = alloc((size_t)N * D_FEAT * 4);
    float* stats = alloc(4 * D_FEAT * 4);                // sum, sumsq, scale, shift
    float* emb   = alloc((size_t)S * D_FEAT * 4);
    float* sc    = stats + 2 * D_FEAT;
    (void)ws_size;

    const int t256 = 256;
    const int gET  = (int)((ET + t256 - 1) / t256);
    const int gN   = (N + t256 - 1) / t256;

    // ---- degree / norm (identical for all 3 layers: compute once) ----
    hipMemsetAsync(dinv, 0, (size_t)N * 4, stream);
    deg_kernel <<<gET, t256, 0, stream>>>(ei, ew, dinv, E, N);
    dinv_kernel<<<gN,  t256, 0, stream>>>(dinv, N);
    norm_kernel<<<gET, t256, 0, stream>>>(ei, ew, dinv, nrm, E, N);

    // ---- 3 GCN layers ----
    const int gGemm  = (N + 15) / 16;                   // 16-row tiles, 8 waves/block
    const int gStats = (N + 63) / 64;
    const int gBn    = (int)(((long long)N * D_FEAT + t256 - 1) / t256);
    const float* hin = x;
    for (int l = 0; l < 3; ++l) {
        gemm128_wmma<<<gGemm, 256, 0, stream>>>(hin, Wl[l], hlin, N);
        hipMemsetAsync(agg,   0, (size_t)N * D_FEAT * 4, stream);
        hipMemsetAsync(stats, 0, 2 * D_FEAT * 4, stream);
        scatter_kernel<<<(unsigned)ET, 128, 0, stream>>>(hlin, ei, nrm, agg, E);
        stats_kernel  <<<gStats, 128, 0, stream>>>(agg, stats, N);
        finalize_stats_kernel<<<1, 128, 0, stream>>>(stats, gl[l], bl[l], sc, N);
        bn_relu_kernel<<<gBn, t256, 0, stream>>>(agg, sc, hbuf, N);
        hin = hbuf;                                      // ping-pong safe: gemm reads before overwrite
    }

    // ---- pooling + classifier ----
    pool_kernel<<<S, 128, 0, stream>>>(hbuf, sg, emb, N);
    mlp_kernel <<<S, 64, 0, stream>>>(emb, Wc1, bc1, Wc2, bc2, out);
    (void)out_size; (void)n_in;
}